// LSTM_38053410243117
// MI455X (gfx1250) — compile-verified
//
#include <hip/hip_runtime.h>
#include <hip/hip_bf16.h>

// Problem constants (from reference): T=512, B=4096, F=16, L=4, H=1.
#define T_LEN 512
#define B_SZ  4096
#define F_IN  16
#define L_N   4

typedef __attribute__((ext_vector_type(2))) float v2f;
typedef __attribute__((ext_vector_type(8))) float v8f;

__device__ __forceinline__ float fast_tanh(float x) {
#if __has_builtin(__builtin_amdgcn_tanhf)
    return __builtin_amdgcn_tanhf(x);          // gfx1250 V_TANH_F32 (1 TRANS op)
#else
    // tanh(x) = 1 - 2/(exp(2x)+1)
    return 1.0f - 2.0f * __builtin_amdgcn_rcpf(__expf(2.0f * x) + 1.0f);
#endif
}

__device__ __forceinline__ float fast_sig(float x) {
    // sigmoid(x) = 0.5*tanh(x/2) + 0.5  -> single TRANS op when V_TANH exists
    return 0.5f * fast_tanh(0.5f * x) + 0.5f;
}

// ---------------------------------------------------------------------------
// Pass 1: gx[m, g] = sum_k x[m,k] * W_ih0[g,k] + (b_ih0[g]+b_hh0[g])
// m = t*B + b flattened (row-major [T,B,F]), so output gx is [T,B,4].
// One wave handles one 16-row M tile via 4 chained V_WMMA_F32_16X16X4_F32.
//   A (16x4 f32): lane L<16 -> row M=L, v0=K0 v1=K1; lane L+16 -> K2/K3.
//   B (4x16 f32): v0 = rows {K0 | K2} striped over lanes, v1 = {K1 | K3}.
//   C/D (16x16):  VGPR j, lanes0-15 -> M=j, lanes16-31 -> M=j+8; N=lane&15.
// N is padded 4->16 (B columns 4..15 are zero; only n<4 is stored).
// ---------------------------------------------------------------------------
__global__ __launch_bounds__(256)
void lstm_gx_wmma(const float* __restrict__ x,      // [T*B, 16]
                  const float* __restrict__ W,      // [4, 16]
                  const float* __restrict__ b_ih,   // [L,4] (layer 0 used)
                  const float* __restrict__ b_hh,
                  float* __restrict__ gx)           // [T*B, 4]
{
    const int lane = threadIdx.x & 31;
    const int wave = threadIdx.x >> 5;
    const int tile = blockIdx.x * 8 + wave;   // 16-row tile
    const int n    = lane & 15;               // output column (gate) / A row
    const int half = lane >> 4;               // K half within a chunk

    // Constant B operand and bias, built once per wave.
    v2f bmat[4];
#pragma unroll
    for (int c = 0; c < 4; ++c) {
        float b0 = 0.f, b1 = 0.f;
        if (n < 4) {
            const int k = 4 * c + 2 * half;
            b0 = W[n * 16 + k];
            b1 = W[n * 16 + k + 1];
        }
        bmat[c].x = b0;
        bmat[c].y = b1;
    }
    const float biasn = (n < 4) ? (b_ih[n] + b_hh[n]) : 0.f;

    const int row = tile * 16 + n;
    const float* xr = x + (size_t)row * F_IN;

    v8f acc;
#pragma unroll
    for (int j = 0; j < 8; ++j) acc[j] = biasn;

#if __has_builtin(__builtin_amdgcn_wmma_f32_16x16x4_f32)
#pragma unroll
    for (int c = 0; c < 4; ++c) {
        v2f a = *(const v2f*)(xr + 4 * c + 2 * half);   // 8B-aligned
        acc = __builtin_amdgcn_wmma_f32_16x16x4_f32(
            /*neg_a=*/false, a, /*neg_b=*/false, bmat[c],
            /*c_mod=*/(short)0, acc, /*reuse_a=*/false, /*reuse_b=*/false);
    }
#else
    // Scalar fallback (same D layout): D[j+8*half][n]
    if (n < 4) {
#pragma unroll
        for (int j = 0; j < 8; ++j) {
            const float* xm = x + ((size_t)tile * 16 + j + 8 * half) * F_IN;
            float s = 0.f;
#pragma unroll
            for (int k = 0; k < F_IN; ++k) s += xm[k] * W[n * 16 + k];
            acc[j] = s + biasn;
        }
    }
#endif

    if (n < 4) {
#pragma unroll
        for (int j = 0; j < 8; ++j) {
            const int m = tile * 16 + j + 8 * half;
            gx[(size_t)m * 4 + n] = acc[j];
        }
    }
}

// ---------------------------------------------------------------------------
// Pass 2: sequential LSTM scan, all 4 layers fused; 1 thread per batch elem.
// gx reads are float4, coalesced across threads, prefetched 8 deep.
// ---------------------------------------------------------------------------
__global__ __launch_bounds__(256)
void lstm_scan(const float4* __restrict__ gxv,       // [T*B] (layer-0 gates+bias)
               const float* __restrict__ Wih_rest,   // [3,4]
               const float* __restrict__ Whh,        // [4,4]
               const float* __restrict__ b_ih,       // [4,4]
               const float* __restrict__ b_hh,       // [4,4]
               float* __restrict__ out)              // [T*B | 4*B | 4*B]
{
    const int b = blockIdx.x * blockDim.x + threadIdx.x;
    if (b >= B_SZ) return;

    float whh[L_N][4], wih[L_N - 1][4], bias[L_N - 1][4];
#pragma unroll
    for (int l = 0; l < L_N; ++l)
#pragma unroll
        for (int g = 0; g < 4; ++g) whh[l][g] = Whh[l * 4 + g];
#pragma unroll
    for (int l = 1; l < L_N; ++l)
#pragma unroll
        for (int g = 0; g < 4; ++g) {
            wih[l - 1][g]  = Wih_rest[(l - 1) * 4 + g];
            bias[l - 1][g] = b_ih[l * 4 + g] + b_hh[l * 4 + g];
        }

    float h[L_N] = {0.f, 0.f, 0.f, 0.f};
    float c[L_N] = {0.f, 0.f, 0.f, 0.f};

    constexpr int PF = 8;                    // prefetch depth (reg ring)
    float4 buf[PF];
#pragma unroll
    for (int k = 0; k < PF; ++k) buf[k] = gxv[(size_t)k * B_SZ + b];

#pragma unroll 8
    for (int t = 0; t < T_LEN; ++t) {
        const float4 g4 = buf[t & (PF - 1)];
        if (t + PF < T_LEN) buf[t & (PF - 1)] = gxv[(size_t)(t + PF) * B_SZ + b];

        // layer 0 (input gates precomputed in gx, biases already folded in)
        float ig = fast_sig (g4.x + h[0] * whh[0][0]);
        float fg = fast_sig (g4.y + h[0] * whh[0][1]);
        float gg = fast_tanh(g4.z + h[0] * whh[0][2]);
        float og = fast_sig (g4.w + h[0] * whh[0][3]);
        c[0] = fg * c[0] + ig * gg;
        h[0] = og * fast_tanh(c[0]);
        float hin = h[0];

#pragma unroll
        for (int l = 1; l < L_N; ++l) {
            const float p0 = bias[l-1][0] + hin * wih[l-1][0] + h[l] * whh[l][0];
            const float p1 = bias[l-1][1] + hin * wih[l-1][1] + h[l] * whh[l][1];
            const float p2 = bias[l-1][2] + hin * wih[l-1][2] + h[l] * whh[l][2];
            const float p3 = bias[l-1][3] + hin * wih[l-1][3] + h[l] * whh[l][3];
            const float i2 = fast_sig(p0), f2 = fast_sig(p1);
            const float g2 = fast_tanh(p2), o2 = fast_sig(p3);
            c[l] = f2 * c[l] + i2 * g2;
            h[l] = o2 * fast_tanh(c[l]);
            hin = h[l];
        }
        out[(size_t)t * B_SZ + b] = hin;     // last-layer hidden sequence
    }

    // hn [L,B,1] then cn [L,B,1]
#pragma unroll
    for (int l = 0; l < L_N; ++l) {
        out[(size_t)T_LEN * B_SZ + (size_t)l * B_SZ + b]                     = h[l];
        out[(size_t)T_LEN * B_SZ + (size_t)L_N * B_SZ + (size_t)l * B_SZ + b] = c[l];
    }
}

extern "C" void kernel_launch(void* const* d_in, const int* in_sizes, int n_in,
                              void* d_out, int out_size, void* d_ws, size_t ws_size,
                              hipStream_t stream) {
    const float* x         = (const float*)d_in[0];  // [T,B,F]
    const float* W_ih0     = (const float*)d_in[1];  // [4,F]
    const float* W_ih_rest = (const float*)d_in[2];  // [L-1,4,1]
    const float* W_hh      = (const float*)d_in[3];  // [L,4,1]
    const float* b_ih      = (const float*)d_in[4];  // [L,4]
    const float* b_hh      = (const float*)d_in[5];  // [L,4]
    float* out = (float*)d_out;
    float* gx  = (float*)d_ws;                       // needs T*B*4*4 = 32 MiB

    // Pass 1: WMMA input projection for layer 0 -> gx [T*B,4]
    const int numTiles = (T_LEN * B_SZ) / 16;        // 131072
    lstm_gx_wmma<<<numTiles / 8, 256, 0, stream>>>(x, W_ih0, b_ih, b_hh, gx);

    // Pass 2: fused 4-layer sequential scan, one thread per batch element
    lstm_scan<<<(B_SZ + 255) / 256, 256, 0, stream>>>(
        (const float4*)gx, W_ih_rest, W_hh, b_ih, b_hh, out);
}